// ReconstructionModule_1812476199713
// MI455X (gfx1250) — compile-verified
//
#include <hip/hip_runtime.h>
#include <stdint.h>

// Problem constants (reference: B=256, N=196, D=768)
#define BATCH 256
#define NPAT  196
#define DDIM  768

// D-chunk staged through LDS per pipeline stage, and padded tile pitch.
// PITCH = 68 floats = 272 bytes: rows stay 16B-aligned for b128 async stores
// into LDS, and the 4-bank skew per row breaks the worst conflicts on the
// p-strided stencil reads.
#define DC    64
#define PITCH 68
#define NCH   (DDIM / DC)   // 12 chunks

// Per-wave async issue count per chunk: 196*16 = 3136 b128 transfers over
// 256 threads -> waves 0-1 issue 13, waves 2-7 issue 12. ASYNCcnt decrements
// oldest-first, so waiting for <= 12 after issuing the next chunk's prefetch
// guarantees the *current* chunk's transfers are all complete.
#define PREF_MIN 12

// -------- CDNA5 async global->LDS helpers (ASYNCcnt path) -----------------
// GLOBAL_LOAD_ASYNC_TO_LDS_B128: per-lane 16B global -> per-lane LDS address.
// The LDS address VGPR holds the byte offset within the workgroup's LDS
// allocation; the low 32 bits of a generic pointer to __shared__ are exactly
// that offset (flat LDS aperture keeps the offset in addr[31:0]).
__device__ __forceinline__ void async_copy_b128(unsigned lds_byte_off,
                                                const float* gsrc) {
  asm volatile("global_load_async_to_lds_b128 %0, %1, off"
               :: "v"(lds_byte_off), "v"(gsrc)
               : "memory");
}

__device__ __forceinline__ void wait_async_le_pref() {
  asm volatile("s_wait_asynccnt 12" ::: "memory");
}
__device__ __forceinline__ void wait_async_zero() {
  asm volatile("s_wait_asynccnt 0" ::: "memory");
}

// ---------------------------------------------------------------------------
// Fused kernel, one block (8 wave32) per batch element:
//   Stage 1: per-column online argmax + max-of-softmax over the 196x196
//            logits (coalesced row reads), confidence = 1/sum exp(x - max).
//            Winner table via LDS atomicMax (largest j wins, matching the
//            reference's later-j-wins scatter).
//   Stage 2: double-buffered ASYNCcnt pipeline: async-gather
//            features[i, winner[p], chunk] into LDS tile k+1 while the
//            masked 3-tap stencil + [N,D]->[D,N] transposed store consumes
//            tile k. Output chunk is a single contiguous coalesced stream.
// ---------------------------------------------------------------------------
__global__ __launch_bounds__(256) void recon_fused_kernel(
    const float* __restrict__ features,   // [B, N, D]
    const float* __restrict__ logits,     // [B, N, N]
    float* __restrict__ img_out,          // [B, D, N]   (N = g*g flattened)
    float* __restrict__ conf_out) {       // [B, N]
  const int i = blockIdx.x;
  const int t = threadIdx.x;

  __shared__ int sw[NPAT];                                // winner table
  __shared__ __align__(16) float tile[2][NPAT * PITCH];   // double buffer

  // ---------------- Stage 1: argmax / confidence / winner ----------------
  if (t < NPAT) sw[t] = -1;
  __syncthreads();

  if (t < NPAT) {
    const float* col = logits + (size_t)i * NPAT * NPAT + t;
    float m = -3.4028235e38f;
    float s = 0.f;
    int   am = 0;
    for (int c = 0; c < NPAT; ++c) {
      float x = col[(size_t)c * NPAT];   // wave reads 32 consecutive floats
      if (x > m) {                       // strict '>' => first max (jnp.argmax)
        s = s * __expf(m - x) + 1.f;
        m = x;
        am = c;
      } else {
        s += __expf(x - m);
      }
    }
    conf_out[(size_t)i * NPAT + t] = 1.f / s;   // max softmax prob
    atomicMax(&sw[am], t);                       // largest j wins collisions
  }
  __syncthreads();

  // ---------------- Stage 2: gather -> stencil -> transpose ----------------
  const float* fbase = features + (size_t)i * NPAT * DDIM;

  auto issue_gather = [&](int ch, float* buf) {
    const int c0 = ch * DC;
    for (int e = t; e < NPAT * (DC / 4); e += 256) {
      const int row  = e >> 4;      // patch index p   (DC/4 = 16 groups/row)
      const int colg = e & 15;      // 4-float group within chunk
      const int wr   = sw[row];
      const int src  = wr < 0 ? 0 : wr;   // dummy row for empty slots (masked)
      const float* g = fbase + (size_t)src * DDIM + c0 + colg * 4;
      const unsigned lds = (unsigned)(size_t)&buf[row * PITCH + colg * 4];
      async_copy_b128(lds, g);
    }
  };

  issue_gather(0, tile[0]);

  for (int ch = 0; ch < NCH; ++ch) {
    if (ch + 1 < NCH) {
      issue_gather(ch + 1, tile[(ch + 1) & 1]);  // prefetch next chunk
      wait_async_le_pref();                      // chunk ch fully landed
    } else {
      wait_async_zero();                         // drain the last chunk
    }
    __syncthreads();   // all waves' async LDS writes for chunk ch visible

    const float* buf = tile[ch & 1];
    float* outc = img_out + ((size_t)i * DDIM + ch * DC) * NPAT;
    for (int l = t; l < DC * NPAT; l += 256) {
      const int dl = l / NPAT;          // d within chunk
      const int p  = l - dl * NPAT;     // patch index
      const float ctr = (sw[p] >= 0) ? buf[p * PITCH + dl] : 0.f;
      float v;
      if (p == 0 || p == NPAT - 1) {
        v = ctr;                        // edges keep original
      } else {
        const float lf = (sw[p - 1] >= 0) ? buf[(p - 1) * PITCH + dl] : 0.f;
        const float rt = (sw[p + 1] >= 0) ? buf[(p + 1) * PITCH + dl] : 0.f;
        v = (lf + ctr + rt) * (1.f / 3.f);
      }
      outc[l] = v;                      // fully coalesced contiguous stream
    }
    __syncthreads();   // protect buffer reused by the next prefetch
  }
}

// ---------------------------------------------------------------------------
extern "C" void kernel_launch(void* const* d_in, const int* in_sizes, int n_in,
                              void* d_out, int out_size, void* d_ws, size_t ws_size,
                              hipStream_t stream) {
  (void)in_sizes; (void)n_in; (void)out_size; (void)d_ws; (void)ws_size;

  const float* features = (const float*)d_in[0];   // [B, N, D] fp32
  const float* logits   = (const float*)d_in[1];   // [B, N, N] fp32

  float* img  = (float*)d_out;                                  // [B, D, N]
  float* conf = (float*)d_out + (size_t)BATCH * DDIM * NPAT;    // [B, N]

  recon_fused_kernel<<<BATCH, 256, 0, stream>>>(features, logits, img, conf);
}